// XEyTransformerLayer_64347200028728
// MI455X (gfx1250) — compile-verified
//
#include <hip/hip_runtime.h>
#include <float.h>

// ---------------------------------------------------------------------------
// CDNA5 (gfx1250) implementation of XEyTransformerLayer forward.
// - All large matmuls use V_WMMA_F32_16X16X4_F32 (wave32, 16x16 C tile).
// - Generic GEMM: 4 waves/block, each wave owns 16 rows x 64 cols (4 N-tiles,
//   one A load feeds 4 WMMAs). B staged per 64-K chunk into LDS with
//   GLOBAL_LOAD_ASYNC_TO_LDS_B128 + S_WAIT_ASYNCCNT (CDNA5 async path).
// - fp32 end-to-end: the layer is HBM/L2-bound, not flop-bound, so fp32 WMMA
//   is free precision.
// ---------------------------------------------------------------------------

typedef __attribute__((ext_vector_type(2))) float v2f;
typedef __attribute__((ext_vector_type(8))) float v8f;

__device__ __forceinline__ v8f wmma4(v2f a, v2f b, v8f c) {
  // 8 args: (neg_a, A, neg_b, B, c_mod, C, reuse_a, reuse_b)
  return __builtin_amdgcn_wmma_f32_16x16x4_f32(false, a, false, b, (short)0, c,
                                               false, false);
}

#define TW 16   // WMMA tile width
#define WPB 4   // waves per block (blockDim.x == 128)

// ---------------------------------------------------------------------------
// Generic GEMM, M%64==0, N%64==0, K%64==0.
// C = epilogue(A@B + bias); MODE: 0 plain, 1 relu, 2 per-row mask.
// Block = 4 waves; block tile = 64 rows x 64 cols; wave tile = 16x64.
// B is staged in 64x64 K-chunks into LDS via async-to-LDS copies.
// ---------------------------------------------------------------------------
template <int MODE>
__global__ __launch_bounds__(128) void gemm64_kernel(
    const float* __restrict__ A, const float* __restrict__ B,
    const float* __restrict__ bias, const int* __restrict__ nmask,
    float* __restrict__ C, int M, int N, int K) {
  __shared__ float sB[64 * 64];
  const int tid = threadIdx.x;
  const int lane = tid & 31;
  const int wave = tid >> 5;
  const int half = lane >> 4, l = lane & 15;
  const int col0 = blockIdx.x * 64;
  const int row0 = blockIdx.y * 64 + wave * TW;

  v8f acc0 = {0.f, 0.f, 0.f, 0.f, 0.f, 0.f, 0.f, 0.f};
  v8f acc1 = acc0, acc2 = acc0, acc3 = acc0;

  const unsigned ldsBase = (unsigned)(unsigned long long)&sB[0];
  const float* ap = A + (size_t)(row0 + l) * K + 2 * half;

  for (int k0 = 0; k0 < K; k0 += 64) {
    // ---- async-stage B[k0:k0+64, col0:col0+64] into LDS (16 KB chunk) ----
    {
      int flat = tid * 4;  // float index into the 64x64 chunk
#pragma unroll
      for (int it = 0; it < 8; ++it) {
        int r = flat >> 6, c = flat & 63;
        const float* gp = B + (size_t)(k0 + r) * N + col0 + c;
        unsigned la = ldsBase + (unsigned)(flat * 4);
        asm volatile("global_load_async_to_lds_b128 %0, %1, off"
                     :: "v"(la), "v"(gp)
                     : "memory");
        flat += 512;
      }
      asm volatile("s_wait_asynccnt 0" ::: "memory");
      __syncthreads();
    }
    if (k0 + 64 < K) __builtin_prefetch(ap + 64, 0, 1);  // warm next A chunk
    // ---- 16 K-steps of 4, one A load feeding 4 column tiles ----
    for (int kk = 0; kk < 64; kk += 4) {
      v2f a;
      a.x = ap[0];
      a.y = ap[1];
      ap += 4;
      const float* bp = &sB[(kk + 2 * half) * 64 + l];
      v2f b;
      b.x = bp[0];   b.y = bp[64];   acc0 = wmma4(a, b, acc0);
      b.x = bp[16];  b.y = bp[80];   acc1 = wmma4(a, b, acc1);
      b.x = bp[32];  b.y = bp[96];   acc2 = wmma4(a, b, acc2);
      b.x = bp[48];  b.y = bp[112];  acc3 = wmma4(a, b, acc3);
    }
    __syncthreads();  // protect LDS before next chunk overwrite
  }

  auto store_tile = [&](const v8f& acc, int tcol) {
    const int col = col0 + tcol + l;
    const float bv = bias[col];
#pragma unroll
    for (int r = 0; r < 8; ++r) {
      int row = row0 + half * 8 + r;
      float v = acc[r] + bv;
      if (MODE == 1) v = fmaxf(v, 0.f);
      if (MODE == 2) v *= (nmask[row] != 0) ? 1.f : 0.f;
      C[(size_t)row * N + col] = v;
    }
  };
  store_tile(acc0, 0);
  store_tile(acc1, 16);
  store_tile(acc2, 32);
  store_tile(acc3, 48);
}

// ---------------------------------------------------------------------------
// Fused E projections + Y construction.
// rows = (b,i,j) flat (65536), K=64 (de), N=256 (dx).
// E1 = (E@Wm + bm)*em ; E2 = (E@Wa + ba)*em
// Y  = Qp[b,j,c]*Kp[b,i,c]*(1/32)*(E1+1) + E2
// ---------------------------------------------------------------------------
__global__ void e_proj_fused_kernel(
    const float* __restrict__ E, const float* __restrict__ Wm,
    const float* __restrict__ bm, const float* __restrict__ Wa,
    const float* __restrict__ ba, const float* __restrict__ Qp,
    const float* __restrict__ Kp, const int* __restrict__ nmask,
    float* __restrict__ Y) {
  const int lane = threadIdx.x & 31;
  const int wave = threadIdx.x >> 5;
  const int col0 = blockIdx.x * TW;
  const int row0 = (blockIdx.y * WPB + wave) * TW;
  const int half = lane >> 4, l = lane & 15;
  v8f acc1 = {0.f, 0.f, 0.f, 0.f, 0.f, 0.f, 0.f, 0.f};
  v8f acc2 = acc1;
  const float* ap = E + (size_t)(row0 + l) * 64 + 2 * half;
  const float* b1 = Wm + (2 * half) * 256 + col0 + l;
  const float* b2 = Wa + (2 * half) * 256 + col0 + l;
  for (int k = 0; k < 64; k += 4) {
    v2f a; a.x = ap[0]; a.y = ap[1];
    v2f x; x.x = b1[0]; x.y = b1[256];
    v2f z; z.x = b2[0]; z.y = b2[256];
    acc1 = wmma4(a, x, acc1);  // shared A operand for both GEMMs
    acc2 = wmma4(a, z, acc2);
    ap += 4;
    b1 += 1024;
    b2 += 1024;
  }
  const int col = col0 + l;
  const float bmv = bm[col], bav = ba[col];
#pragma unroll
  for (int r = 0; r < 8; ++r) {
    int row = row0 + half * 8 + r;
    int b = row >> 14, i = (row >> 7) & 127, j = row & 127;
    float em = (nmask[b * 128 + i] != 0 && nmask[b * 128 + j] != 0) ? 1.f : 0.f;
    float e1 = (acc1[r] + bmv) * em;
    float e2 = (acc2[r] + bav) * em;
    float q = Qp[(size_t)(b * 128 + j) * 256 + col];
    float kk = Kp[(size_t)(b * 128 + i) * 256 + col];
    Y[(size_t)row * 256 + col] = q * kk * (1.f / 32.f) * (e1 + 1.f) + e2;
  }
}

// ---------------------------------------------------------------------------
// e_out GEMM with y-modulation fused into the A fetch:
// newE = ((yeAdd + (yeMul+1)*Y) @ W + bias) * em
// rows = 65536, K=256, N=64.
// ---------------------------------------------------------------------------
__global__ void e_out_kernel(const float* __restrict__ Y,
                             const float* __restrict__ W,
                             const float* __restrict__ bias,
                             const float* __restrict__ yeAdd,
                             const float* __restrict__ yeMul,
                             const int* __restrict__ nmask,
                             float* __restrict__ newE) {
  const int lane = threadIdx.x & 31;
  const int wave = threadIdx.x >> 5;
  const int col0 = blockIdx.x * TW;
  const int row0 = (blockIdx.y * WPB + wave) * TW;
  const int half = lane >> 4, l = lane & 15;
  const int arow = row0 + l;
  const int bA = arow >> 14;  // batch constant within a 16-row tile
  const float* yp = Y + (size_t)arow * 256 + 2 * half;
  const float* ya = yeAdd + bA * 256 + 2 * half;
  const float* ym = yeMul + bA * 256 + 2 * half;
  const float* wp = W + (2 * half) * 64 + col0 + l;
  v8f acc = {0.f, 0.f, 0.f, 0.f, 0.f, 0.f, 0.f, 0.f};
  for (int k = 0; k < 256; k += 4) {
    v2f a;
    a.x = ya[0] + (ym[0] + 1.f) * yp[0];
    a.y = ya[1] + (ym[1] + 1.f) * yp[1];
    v2f b; b.x = wp[0]; b.y = wp[64];
    acc = wmma4(a, b, acc);
    yp += 4; ya += 4; ym += 4;
    wp += 256;
  }
  const int col = col0 + l;
  const float bv = bias[col];
#pragma unroll
  for (int r = 0; r < 8; ++r) {
    int row = row0 + half * 8 + r;
    int b = row >> 14, i = (row >> 7) & 127, j = row & 127;
    float em = (nmask[b * 128 + i] != 0 && nmask[b * 128 + j] != 0) ? 1.f : 0.f;
    newE[(size_t)row * 64 + col] = (acc[r] + bv) * em;
  }
}

// ---------------------------------------------------------------------------
// Per-(b,i,channel) online softmax over j + weighted sum with V + y-modulation
// modX[b,i,c] = yxAdd[b,c] + (yxMul[b,c]+1) * sum_j softmax_j(Y[b,i,j,c]) V[b,j,c]
// ---------------------------------------------------------------------------
__global__ void attn_kernel(const float* __restrict__ Y,
                            const float* __restrict__ Vp,
                            const int* __restrict__ nmask,
                            const float* __restrict__ yxAdd,
                            const float* __restrict__ yxMul,
                            float* __restrict__ modX) {
  int idx = blockIdx.x * blockDim.x + threadIdx.x;
  if (idx >= 4 * 128 * 256) return;
  const int c = idx & 255;
  const int bi = idx >> 8;  // b*128 + i
  const int b = bi >> 7;
  const bool valid = nmask[bi] != 0;  // sm masks axis 1 (the i index)
  const float* yp = Y + (size_t)bi * 128 * 256 + c;
  const float* vp = Vp + (size_t)b * 128 * 256 + c;
  float m = -FLT_MAX, s = 0.f, wv = 0.f;
  for (int j = 0; j < 128; ++j) {
    float v = valid ? yp[(size_t)j * 256] : -FLT_MAX;
    float vecv = vp[(size_t)j * 256];
    float nm = fmaxf(m, v);
    float sc = __expf(m - nm);
    float e = __expf(v - nm);
    s = s * sc + e;
    wv = wv * sc + e * vecv;
    m = nm;
  }
  float w = wv / s;
  modX[idx] = yxAdd[b * 256 + c] + (yxMul[b * 256 + c] + 1.f) * w;
}

// ---------------------------------------------------------------------------
// LayerNorm(x + r) * gamma + beta. One block per row, blockDim == d (64/256).
// Safe in-place (out may alias x): all loads happen before any store.
// ---------------------------------------------------------------------------
__global__ void ln_residual_kernel(const float* __restrict__ x,
                                   const float* __restrict__ r,
                                   const float* __restrict__ gamma,
                                   const float* __restrict__ beta,
                                   float* __restrict__ out, int d) {
  __shared__ float red[256];
  const int t = threadIdx.x;
  const size_t row = blockIdx.x;
  float v = x[row * d + t] + r[row * d + t];
  red[t] = v;
  __syncthreads();
  for (int s = d >> 1; s > 0; s >>= 1) {
    if (t < s) red[t] += red[t + s];
    __syncthreads();
  }
  float mean = red[0] / (float)d;
  __syncthreads();
  float dv = v - mean;
  red[t] = dv * dv;
  __syncthreads();
  for (int s = d >> 1; s > 0; s >>= 1) {
    if (t < s) red[t] += red[t + s];
    __syncthreads();
  }
  float var = red[0] / (float)d;
  out[row * d + t] = dv * rsqrtf(var + 1e-5f) * gamma[t] + beta[t];
}

// ---------------------------------------------------------------------------
// zx = [mean_i X, min_i X, max_i X] per (b,c): (4,768)
// ---------------------------------------------------------------------------
__global__ void x_stats_kernel(const float* __restrict__ X,
                               float* __restrict__ zx) {
  int idx = blockIdx.x * blockDim.x + threadIdx.x;
  if (idx >= 4 * 256) return;
  int b = idx >> 8, c = idx & 255;
  float s = 0.f, mn = FLT_MAX, mx = -FLT_MAX;
  for (int i = 0; i < 128; ++i) {
    float v = X[((size_t)(b * 128 + i)) * 256 + c];
    s += v;
    mn = fminf(mn, v);
    mx = fmaxf(mx, v);
  }
  zx[b * 768 + c] = s / 128.f;
  zx[b * 768 + 256 + c] = mn;
  zx[b * 768 + 512 + c] = mx;
}

// ---------------------------------------------------------------------------
// ze = [mean_{i,j} E, min, max] per (b,c): (4,192). One block per (b,c).
// ---------------------------------------------------------------------------
__global__ void e_stats_kernel(const float* __restrict__ E,
                               float* __restrict__ ze) {
  __shared__ float ss[256], smn[256], smx[256];
  const int b = blockIdx.x >> 6, c = blockIdx.x & 63, t = threadIdx.x;
  float s = 0.f, mn = FLT_MAX, mx = -FLT_MAX;
  for (int k = t; k < 16384; k += 256) {
    float v = E[((size_t)b * 16384 + k) * 64 + c];
    s += v;
    mn = fminf(mn, v);
    mx = fmaxf(mx, v);
  }
  ss[t] = s; smn[t] = mn; smx[t] = mx;
  __syncthreads();
  for (int st = 128; st > 0; st >>= 1) {
    if (t < st) {
      ss[t] += ss[t + st];
      smn[t] = fminf(smn[t], smn[t + st]);
      smx[t] = fmaxf(smx[t], smx[t + st]);
    }
    __syncthreads();
  }
  if (t == 0) {
    ze[b * 192 + c] = ss[0] / 16384.f;
    ze[b * 192 + 64 + c] = smn[0];
    ze[b * 192 + 128 + c] = smx[0];
  }
}

// ---------------------------------------------------------------------------
// Naive dense for the tiny y-path (M=4): out [+]= relu?(A@W + bias)
// ---------------------------------------------------------------------------
__global__ void small_dense_kernel(const float* __restrict__ A,
                                   const float* __restrict__ W,
                                   const float* __restrict__ bias,
                                   float* __restrict__ out, int M, int K, int N,
                                   int relu, int accum) {
  int idx = blockIdx.x * blockDim.x + threadIdx.x;
  if (idx >= M * N) return;
  int m = idx / N, n = idx % N;
  float acc = bias[n];
  const float* a = A + (size_t)m * K;
  for (int k = 0; k < K; ++k) acc += a[k] * W[(size_t)k * N + n];
  if (relu) acc = fmaxf(acc, 0.f);
  if (accum) acc += out[idx];
  out[idx] = acc;
}

// ---------------------------------------------------------------------------
// Host launch
// ---------------------------------------------------------------------------
extern "C" void kernel_launch(void* const* d_in, const int* in_sizes, int n_in,
                              void* d_out, int out_size, void* d_ws,
                              size_t ws_size, hipStream_t stream) {
  (void)in_sizes; (void)n_in; (void)out_size; (void)ws_size;
  const float* X = (const float*)d_in[0];  // (4,128,256)
  const float* E = (const float*)d_in[1];  // (4,128,128,64)
  const float* y = (const float*)d_in[2];  // (4,64)
  const int* nmask = (const int*)d_in[3];  // (4,128)
#define PW(i) ((const float*)d_in[i])
  // params: 4:q_w 5:q_b 6:k_w 7:k_b 8:v_w 9:v_b 10:e_mul_w 11:e_mul_b
  // 12:e_add_w 13:e_add_b 14:y_e_mul_w 15:y_e_mul_b 16:y_e_add_w 17:y_e_add_b
  // 18:y_x_mul_w 19:y_x_mul_b 20:y_x_add_w 21:y_x_add_b 22:x_out_w 23:x_out_b
  // 24:e_out_w 25:e_out_b 26:y_y_w 27:y_y_b 28:x_to_y_w 29:x_to_y_b
  // 30:e_to_y_w 31:e_to_y_b 32:y_out1_w 33:y_out1_b 34:y_out2_w 35:y_out2_b
  // 36:linX1_w 37:linX1_b 38:linX2_w 39:linX2_b 40:linE1_w 41:linE1_b
  // 42:linE2_w 43:linE2_b 44:lin_y1_w 45:lin_y1_b 46:lin_y2_w 47:lin_y2_b
  // 48:normX1_s 49:normX1_b 50:normX2_s 51:normX2_b 52:normE1_s 53:normE1_b
  // 54:normE2_s 55:normE2_b 56:norm_y1_s 57:norm_y1_b 58:norm_y2_s 59:norm_y2_b

  float* w = (float*)d_ws;
  const size_t OFF_QP = 0;           // 512*256
  const size_t OFF_KP = 131072;      // 512*256
  const size_t OFF_VP = 262144;      // 512*256
  const size_t OFF_MODX = 393216;    // 512*256 (softmax output, modulated)
  const size_t OFF_XR = 524288;      // 512*256
  const size_t OFF_XF1 = 655360;     // 512*1024
  const size_t OFF_XF = 1179648;     // 512*256 (newX, later Xf)
  const size_t OFF_YEMUL = 1310720;  // 4*256
  const size_t OFF_YEADD = 1311744;  // 4*256
  const size_t OFF_YXMUL = 1312768;  // 4*256
  const size_t OFF_YXADD = 1313792;  // 4*256
  const size_t OFF_ZX = 1314816;     // 4*768
  const size_t OFF_ZE = 1317888;     // 4*192
  const size_t OFF_YL = 1318656;     // 4*64 (new_y accumulator)
  const size_t OFF_YH = 1318912;     // 4*64 temp
  const size_t OFF_YR = 1319168;     // 4*64 (yr)
  const size_t OFF_YT = 1319424;     // 4*256 temp
  const size_t OFF_NEWE = 1320448;   // 65536*64 (newE; reused as Ef)
  const size_t OFF_Y = 5514752;      // 65536*256 (Y; reused as Ef1)
  // total = 22291968 floats ~= 85 MB

  float* Xo = (float*)d_out;  // 4*128*256
  float* Eo = Xo + 131072;    // 4*128*128*64
  float* yo = Eo + 4194304;   // 4*64

  dim3 blk(128);

  // --- y modulation vectors (tiny GEMMs) ---
  small_dense_kernel<<<dim3(4), dim3(256), 0, stream>>>(y, PW(14), PW(15), w + OFF_YEMUL, 4, 64, 256, 0, 0);
  small_dense_kernel<<<dim3(4), dim3(256), 0, stream>>>(y, PW(16), PW(17), w + OFF_YEADD, 4, 64, 256, 0, 0);
  small_dense_kernel<<<dim3(4), dim3(256), 0, stream>>>(y, PW(18), PW(19), w + OFF_YXMUL, 4, 64, 256, 0, 0);
  small_dense_kernel<<<dim3(4), dim3(256), 0, stream>>>(y, PW(20), PW(21), w + OFF_YXADD, 4, 64, 256, 0, 0);

  // --- Q/K/V projections: (X@W + b) * xm, 512x256x256 (WMMA + async LDS) ---
  gemm64_kernel<2><<<dim3(4, 8), blk, 0, stream>>>(X, PW(4), PW(5), nmask, w + OFF_QP, 512, 256, 256);
  gemm64_kernel<2><<<dim3(4, 8), blk, 0, stream>>>(X, PW(6), PW(7), nmask, w + OFF_KP, 512, 256, 256);
  gemm64_kernel<2><<<dim3(4, 8), blk, 0, stream>>>(X, PW(8), PW(9), nmask, w + OFF_VP, 512, 256, 256);

  // --- fused E1/E2 GEMMs + Y tensor (WMMA, 65536x64x256, dual accum) ---
  e_proj_fused_kernel<<<dim3(16, 1024), blk, 0, stream>>>(
      E, PW(10), PW(11), PW(12), PW(13), w + OFF_QP, w + OFF_KP, nmask, w + OFF_Y);

  // --- newE = ((yeAdd + (yeMul+1)*Y) @ e_out + b) * em (WMMA, 65536x256x64) ---
  e_out_kernel<<<dim3(4, 1024), blk, 0, stream>>>(
      w + OFF_Y, PW(24), PW(25), w + OFF_YEADD, w + OFF_YEMUL, nmask, w + OFF_NEWE);

  // --- online softmax over j + V weighting + y-modulation ---
  attn_kernel<<<dim3(512), dim3(256), 0, stream>>>(
      w + OFF_Y, w + OFF_VP, nmask, w + OFF_YXADD, w + OFF_YXMUL, w + OFF_MODX);

  // --- newX = (modX @ x_out + b) * xm (WMMA, 512x256x256) ---
  gemm64_kernel<2><<<dim3(4, 8), blk, 0, stream>>>(
      w + OFF_MODX, PW(22), PW(23), nmask, w + OFF_XF, 512, 256, 256);

  // --- global pooled stats ---
  x_stats_kernel<<<dim3(4), dim3(256), 0, stream>>>(X, w + OFF_ZX);
  e_stats_kernel<<<dim3(256), dim3(256), 0, stream>>>(E, w + OFF_ZE);

  // --- y path (tiny) ---
  small_dense_kernel<<<dim3(1), dim3(256), 0, stream>>>(y, PW(26), PW(27), w + OFF_YL, 4, 64, 64, 0, 0);
  small_dense_kernel<<<dim3(1), dim3(256), 0, stream>>>(w + OFF_ZX, PW(28), PW(29), w + OFF_YL, 4, 768, 64, 0, 1);
  small_dense_kernel<<<dim3(1), dim3(256), 0, stream>>>(w + OFF_ZE, PW(30), PW(31), w + OFF_YL, 4, 192, 64, 0, 1);
  small_dense_kernel<<<dim3(1), dim3(256), 0, stream>>>(w + OFF_YL, PW(32), PW(33), w + OFF_YH, 4, 64, 64, 1, 0);
  small_dense_kernel<<<dim3(1), dim3(256), 0, stream>>>(w + OFF_YH, PW(34), PW(35), w + OFF_YT, 4, 64, 64, 0, 0);
  ln_residual_kernel<<<dim3(4), dim3(64), 0, stream>>>(y, w + OFF_YT, PW(56), PW(57), w + OFF_YR, 64);
  small_dense_kernel<<<dim3(4), dim3(256), 0, stream>>>(w + OFF_YR, PW(44), PW(45), w + OFF_YT, 4, 64, 256, 1, 0);
  small_dense_kernel<<<dim3(1), dim3(256), 0, stream>>>(w + OFF_YT, PW(46), PW(47), w + OFF_YH, 4, 256, 64, 0, 0);
  ln_residual_kernel<<<dim3(4), dim3(64), 0, stream>>>(w + OFF_YR, w + OFF_YH, PW(58), PW(59), yo, 64);

  // --- X path: Xr = LN(X + newX); FF; Xo = LN(Xr + Xf) ---
  ln_residual_kernel<<<dim3(512), dim3(256), 0, stream>>>(X, w + OFF_XF, PW(48), PW(49), w + OFF_XR, 256);
  gemm64_kernel<1><<<dim3(16, 8), blk, 0, stream>>>(w + OFF_XR, PW(36), PW(37), nullptr, w + OFF_XF1, 512, 1024, 256);
  gemm64_kernel<0><<<dim3(4, 8), blk, 0, stream>>>(w + OFF_XF1, PW(38), PW(39), nullptr, w + OFF_XF, 512, 256, 1024);
  ln_residual_kernel<<<dim3(512), dim3(256), 0, stream>>>(w + OFF_XR, w + OFF_XF, PW(50), PW(51), Xo, 256);

  // --- E path: Er = LN(E + newE); FF; Eo = LN(Er + Ef) ---
  // Er lives in the Eo output region; final LN runs safely in place.
  ln_residual_kernel<<<dim3(65536), dim3(64), 0, stream>>>(E, w + OFF_NEWE, PW(52), PW(53), Eo, 64);
  // Ef1 reuses the (now dead) Y region; Ef reuses the newE region.
  gemm64_kernel<1><<<dim3(4, 1024), blk, 0, stream>>>(Eo, PW(40), PW(41), nullptr, w + OFF_Y, 65536, 256, 64);
  gemm64_kernel<0><<<dim3(1, 1024), blk, 0, stream>>>(w + OFF_Y, PW(42), PW(43), nullptr, w + OFF_NEWE, 65536, 64, 256);
  ln_residual_kernel<<<dim3(65536), dim3(64), 0, stream>>>(Eo, w + OFF_NEWE, PW(54), PW(55), Eo, 64);
#undef PW
}